// LinearReceiver_19018115187270
// MI455X (gfx1250) — compile-verified
//
#include <hip/hip_runtime.h>
#include <cstdint>
#include <cstddef>

// Problem constants (from reference)
#define VOCAB   1024
#define SLEN    32
#define NBATCH  4096
#define NOUT    1024
#define ROWLEN  (VOCAB * SLEN)      // 32768 floats per W row

// Kernel tiling
#define OTILE   2                   // output rows per block
#define TPB     256                 // threads per block
#define CHUNK   256                 // batches per x-staging chunk
#define NPAIR   (SLEN / 2)          // 16 int2 token-pairs per batch
#define XSTR    257                 // LDS stride (in int2 units); odd -> conflict-free

// LDS layout (dynamic shared, single allocation):
//   [0, 262144)            : 2 planes of W row  (2 * 32768 floats)
//   [262144, 262144+32896) : xpairs[16][XSTR] int2 (tokens for (s=2m, 2m+1))
#define LDS_WBYTES  (OTILE * ROWLEN * 4)
#define LDS_TOTAL   (LDS_WBYTES + NPAIR * XSTR * 8)

typedef float v2f __attribute__((ext_vector_type(2)));

__global__ __launch_bounds__(TPB) void linrecv_gather_kernel(
    const int* __restrict__ x,      // [4096, 32]
    const float* __restrict__ W,    // [1024, 32768]
    const float* __restrict__ bias, // [1024]
    float* __restrict__ out)        // [4096, 1024]
{
    extern __shared__ char smem[];
    float* wrow = (float*)smem;                         // 2*32768 floats
    const int2* xpairs = (const int2*)(smem + LDS_WBYTES);

    const int tid = threadIdx.x;
    const int bo  = blockIdx.x;                         // output pair index (0..511)

    // Low 32 bits of a generic LDS pointer == byte offset within LDS.
    const unsigned lds0 = (unsigned)(uintptr_t)wrow;
    const unsigned ldsx = lds0 + (unsigned)LDS_WBYTES;

    // ---- Stage two W rows (2 x 128 KB) into LDS via CDNA5 async global->LDS ----
#pragma unroll
    for (int j = 0; j < OTILE; ++j) {
        const char* src = (const char*)(W + (size_t)(OTILE * bo + j) * ROWLEN);
        const unsigned ldsPlane = lds0 + (unsigned)j * (ROWLEN * 4u);
#pragma unroll
        for (int k = 0; k < (ROWLEN * 4) / 16 / TPB; ++k) {   // 32 x 16B per thread
            unsigned goff = (unsigned)(tid + k * TPB) * 16u;
            unsigned ldst = ldsPlane + goff;
            asm volatile("global_load_async_to_lds_b128 %0, %1, %2"
                         :: "v"(ldst), "v"(goff), "s"(src)
                         : "memory");
        }
    }

    const float b0 = bias[OTILE * bo + 0];
    const float b1 = bias[OTILE * bo + 1];

    asm volatile("s_wait_asynccnt 0" ::: "memory");
    __syncthreads();

    // ---- Batch chunks: async transpose-scatter of tokens, then gather-sum ----
    for (int chunk = 0; chunk < NBATCH / CHUNK; ++chunk) {
        // Each lane copies int2 token-pairs with a per-lane scattered LDS
        // destination: global pair q = r*16 + m  ->  LDS xpairs[m][r].
        const char* xsrc = (const char*)(x + (size_t)chunk * CHUNK * SLEN);
#pragma unroll
        for (int k = 0; k < (CHUNK * NPAIR) / TPB; ++k) {     // 16 b64 per thread
            unsigned q    = (unsigned)tid + (unsigned)k * TPB;
            unsigned goff = q * 8u;
            unsigned ldst = ldsx + ((q & (NPAIR - 1u)) * (unsigned)XSTR + (q >> 4)) * 8u;
            asm volatile("global_load_async_to_lds_b64 %0, %1, %2"
                         :: "v"(ldst), "v"(goff), "s"(xsrc)
                         : "memory");
        }
        asm volatile("s_wait_asynccnt 0" ::: "memory");
        __syncthreads();

        // Each thread: one batch of this chunk, both output rows.
        const int bi = chunk * CHUNK + tid;
        float a0 = b0, a1 = b1;
#pragma unroll
        for (int m = 0; m < NPAIR; ++m) {
            int2 tp = xpairs[m * XSTR + tid];          // conflict-free b64 read
            int off0 = ((2 * m + 0) << 10) + tp.x;     // s*1024 + token
            int off1 = ((2 * m + 1) << 10) + tp.y;
            a0 += wrow[off0];                          // random gather, plane 0
            a1 += wrow[ROWLEN + off0];                 // random gather, plane 1
            a0 += wrow[off1];
            a1 += wrow[ROWLEN + off1];
        }
        v2f o2 = {a0, a1};
        __builtin_nontemporal_store(o2, (v2f*)(out + (size_t)bi * NOUT + OTILE * bo));
        __syncthreads();                               // xpairs reused next chunk
    }
}

extern "C" void kernel_launch(void* const* d_in, const int* in_sizes, int n_in,
                              void* d_out, int out_size, void* d_ws, size_t ws_size,
                              hipStream_t stream) {
    const int*   x    = (const int*)d_in[0];    // [4096, 32] int32
    const float* W    = (const float*)d_in[1];  // [1024, 32768] f32
    const float* bias = (const float*)d_in[2];  // [1024] f32
    float* out = (float*)d_out;

    // Tuple outputs 2 and 3 are zeros[4096] each, concatenated after result.
    hipMemsetAsync(out + (size_t)NBATCH * NOUT, 0, 2u * NBATCH * sizeof(float), stream);

    // 295,040 bytes of dynamic LDS (<= 320 KB per WGP).
    (void)hipFuncSetAttribute((const void*)linrecv_gather_kernel,
                              hipFuncAttributeMaxDynamicSharedMemorySize,
                              (int)LDS_TOTAL);

    dim3 grid(NOUT / OTILE);   // 512 blocks, one per output-row pair
    dim3 block(TPB);
    linrecv_gather_kernel<<<grid, block, LDS_TOTAL, stream>>>(x, W, bias, out);
}